// xSchNet_89713276879268
// MI455X (gfx1250) — compile-verified
//
#include <hip/hip_runtime.h>
#include <math.h>

#define BATCHN 16384
#define NWAVES 8          // wave32 waves per 256-thread block; one molecule per wave
#define ND 18
#define LDA 19            // row pitch to spread LDS banks
#define NSWEEP 12

typedef __attribute__((ext_vector_type(2))) float v2f;
typedef __attribute__((ext_vector_type(8))) float v8f;

__device__ __forceinline__ void wsync() {
  // wave32 executes in lockstep and DS ops are in-order per wave; this only
  // stops the compiler from reordering LDS traffic across the sync point.
  __builtin_amdgcn_wave_barrier();
}

__device__ __forceinline__ float sspf(float x) {  // softplus(x) - ln2, overflow-safe
  float sp = (x > 0.f) ? (x + log1pf(expf(-x))) : log1pf(expf(x));
  return sp - 0.69314718055994530942f;
}

__global__ __launch_bounds__(256)
void xschnet_fused(
    const float* __restrict__ pos, const float* __restrict__ hg,
    const float* __restrict__ emb, const float* __restrict__ Win,
    const float* __restrict__ Wf1, const float* __restrict__ bf1,
    const float* __restrict__ Wf2, const float* __restrict__ bf2,
    const float* __restrict__ Wo1, const float* __restrict__ bo1,
    const float* __restrict__ Wo2, const float* __restrict__ bo2,
    const float* __restrict__ Wt,  const float* __restrict__ bt,
    const float* __restrict__ Won1,const float* __restrict__ bon1,
    const float* __restrict__ Won2,const float* __restrict__ bon2,
    const float* __restrict__ Wof1,const float* __restrict__ bof1,
    const float* __restrict__ Wof2,const float* __restrict__ bof2,
    const float* __restrict__ Fd,  const float* __restrict__ maskg,
    float* __restrict__ out)
{
  __shared__ float sA [NWAVES][ND*LDA];   // working matrix -> diagonalized
  __shared__ float sU [NWAVES][ND*LDA];   // accumulated eigenvectors
  __shared__ float sH [NWAVES][ND*LDA];   // h_new copy
  __shared__ float sOn[NWAVES][81];       // 2*on81 staging
  __shared__ float sOf[NWAVES][81];       // off81 staging
  __shared__ float sE [NWAVES][ND];       // sorted eigenvalues
  __shared__ int   sCol[NWAVES][8];       // source column for sorted rank < 7
  __shared__ float sUo[NWAVES][ND*8];     // Uocc, padded to 8 cols (col7 = 0)

  const int w    = threadIdx.x >> 5;
  const int lane = threadIdx.x & 31;
  const int b    = blockIdx.x * NWAVES + w;
  if (b >= BATCHN) return;                 // uniform per wave (never taken: 2048*8)

  float* A = sA[w];
  float* U = sU[w];
  float* H = sH[w];

  // ---------------- geometry (redundant across lanes; trivial) ----------------
  const float dx = pos[b*6+0] - pos[b*6+3];
  const float dy = pos[b*6+1] - pos[b*6+4];
  const float dz = pos[b*6+2] - pos[b*6+5];
  const float d  = sqrtf(dx*dx + dy*dy + dz*dz);

  const float width = 10.f/19.f;           // linspace(0,10,20) step
  const float coef  = -0.5f/(width*width);
  const float rcut  = (d < 10.f) ? 0.5f*(cosf(d*0.31415926535897932385f) + 1.f) : 0.f;

  // -------- tiny feature chain (NS=5). Both atoms use embedding[7]. ----------
  float x0[5], xf[5];
#pragma unroll
  for (int i = 0; i < 5; ++i) x0[i] = emb[7*5 + i];
#pragma unroll
  for (int s = 0; s < 5; ++s) {
    float a = 0.f;
#pragma unroll
    for (int i = 0; i < 5; ++i) a += x0[i]*Win[i*5+s];
    xf[s] = a;
  }

  float fk[20];
#pragma unroll
  for (int k = 0; k < 20; ++k) { float dd = d - width*(float)k; fk[k] = expf(coef*dd*dd); }

  float h1[5];
#pragma unroll
  for (int s = 0; s < 5; ++s) {
    float a = bf1[s];
#pragma unroll
    for (int k = 0; k < 20; ++k) a += fk[k]*Wf1[k*5+s];
    h1[s] = sspf(a);
  }
  float Wij[5];
#pragma unroll
  for (int s = 0; s < 5; ++s) {
    float a = bf2[s];
#pragma unroll
    for (int t = 0; t < 5; ++t) a += h1[t]*Wf2[t*5+s];
    Wij[s] = a * rcut;
  }
  // agg is identical for both atoms (xf[atom0]==xf[atom1])
  float agg[5];
#pragma unroll
  for (int s = 0; s < 5; ++s) agg[s] = xf[s]*Wij[s];

  float t5[5];
#pragma unroll
  for (int s = 0; s < 5; ++s) {
    float a = bo1[s];
#pragma unroll
    for (int i = 0; i < 5; ++i) a += agg[i]*Wo1[i*5+s];
    t5[s] = sspf(a);
  }
  float xa[5];
#pragma unroll
  for (int s = 0; s < 5; ++s) {
    float a = bo2[s];
#pragma unroll
    for (int t = 0; t < 5; ++t) a += t5[t]*Wo2[t*5+s];
    xa[s] = x0[s] + a;
  }
  float xt[5];
#pragma unroll
  for (int s = 0; s < 5; ++s) {
    float a = bt[s];
#pragma unroll
    for (int i = 0; i < 5; ++i) a += xa[i]*Wt[i*5+s];
    xt[s] = a;
  }
  // all four outer products p[k,l] are identical: p5[i] = xt[i]^2 ; scale == 0
  float p5[5];
#pragma unroll
  for (int i = 0; i < 5; ++i) p5[i] = xt[i]*xt[i];

  float hon[5], hof[5];
#pragma unroll
  for (int s = 0; s < 5; ++s) {
    float a = bon1[s], c2 = bof1[s];
#pragma unroll
    for (int i = 0; i < 5; ++i) { a += p5[i]*Won1[i*5+s]; c2 += p5[i]*Wof1[i*5+s]; }
    hon[s] = sspf(a);
    hof[s] = sspf(c2);
  }

  // 81-wide output layers, distributed across lanes
  for (int j = lane; j < 81; j += 32) {
    float von = bon2[j], vof = bof2[j];
#pragma unroll
    for (int t = 0; t < 5; ++t) { von += hon[t]*Won2[t*81+j]; vof += hof[t]*Wof2[t*81+j]; }
    sOn[w][j] = 2.f*von;   // F_on_raw = sum over l of two identical blocks
    sOf[w][j] = vof;       // scale==0, so Foff_final = 0.5*(of + of^T)
  }
  wsync();

  // ---------------- assemble F_new (18x18), write output, seed Jacobi --------
  for (int idx = lane; idx < 324; idx += 32) {
    int r = idx / 18, c = idx - r*18;
    float J;
    if (r < 9) {
      if (c < 9) { J = 0.5f*(sOn[w][r*9+c] + sOn[w][c*9+r]); }
      else       { int c2 = c-9; J = 0.5f*(sOf[w][c2*9+r] + sOf[w][r*9+c2]); }
    } else {
      int r2 = r-9;
      if (c < 9) { J = 0.5f*(sOf[w][r2*9+c] + sOf[w][c*9+r2]); }
      else       { int c2 = c-9; J = 0.5f*(sOn[w][r2*9+c2] + sOn[w][c2*9+r2]); }
    }
    float hv = hg[(size_t)b*324 + idx];
    H[r*LDA+c] = hv;
    float Fv = ((fabsf(hv) > 1e-7f) ? maskg[idx]*J : 0.f) + Fd[idx];
    A[r*LDA+c] = Fv;
    U[r*LDA+c] = (r == c) ? 1.f : 0.f;
    out[(size_t)b*324 + idx] = Fv;
  }
  wsync();

  // ---------------- cyclic Jacobi eigensolver, one wave per matrix -----------
  for (int sweep = 0; sweep < NSWEEP; ++sweep) {
    for (int p = 0; p < ND-1; ++p) {
      for (int q = p+1; q < ND; ++q) {
        float apq = A[p*LDA+q];                       // LDS broadcast read
        if (fabsf(apq) > 1e-12f) {                    // uniform branch
          float app = A[p*LDA+p], aqq = A[q*LDA+q];
          float th  = 0.5f*(aqq - app)/apq;
          float t   = copysignf(1.f, th)/(fabsf(th) + sqrtf(th*th + 1.f));
          float cth = 1.f/sqrtf(t*t + 1.f);
          float sth = t*cth;
          int k = lane;
          if (k < ND) {                               // B = A*J (columns)
            float akp = A[k*LDA+p], akq = A[k*LDA+q];
            A[k*LDA+p] = cth*akp - sth*akq;
            A[k*LDA+q] = sth*akp + cth*akq;
          }
          wsync();
          if (k < ND) {                               // A' = J^T*B (rows)
            float apk = A[p*LDA+k], aqk = A[q*LDA+k];
            A[p*LDA+k] = cth*apk - sth*aqk;
            A[q*LDA+k] = sth*apk + cth*aqk;
          }
          if (k < ND) {                               // U = U*J
            float ukp = U[k*LDA+p], ukq = U[k*LDA+q];
            U[k*LDA+p] = cth*ukp - sth*ukq;
            U[k*LDA+q] = sth*ukp + cth*ukq;
          }
          wsync();
        }
      }
    }
  }

  // ---------------- sort eigenvalues ascending (parallel ranking) ------------
  if (lane < ND) {
    float ev = A[lane*LDA+lane];
    int rank = 0;
#pragma unroll
    for (int j = 0; j < ND; ++j) {
      float evj = A[j*LDA+j];
      rank += ((evj < ev) || (evj == ev && j < lane)) ? 1 : 0;
    }
    sE[w][rank] = ev;
    if (rank < 7) sCol[w][rank] = lane;
  }
  wsync();

  // gather Uocc (18x7) into padded 18x8 (col 7 = 0)
  for (int idx = lane; idx < ND*8; idx += 32) {
    int r = idx >> 3, cpos = idx & 7;
    sUo[w][idx] = (cpos < 7) ? U[r*LDA + sCol[w][cpos]] : 0.f;
  }
  wsync();

  if (lane < ND) out[(size_t)BATCHN*324 + (size_t)b*18 + lane] = sE[w][lane];

  // ------- E_h = sum_{p,q} h[p,q] * (Uocc Uocc^T)[p,q] -----------------------
  // 16x16 block via two fp32 WMMAs (K = 7 padded to 8, split 4+4).
  // For A*A^T the A and B operand register layouts coincide (ISA 7.12.2), so
  // the same VGPR pair is passed as both SRC0 and SRC1. EXEC is all-1s here.
  const int lane16 = lane & 15;
  const int hi     = lane >> 4;
  const float* Uo  = sUo[w];
  v2f a0, a1;
  a0.x = Uo[lane16*8 + (hi ? 2 : 0)];
  a0.y = Uo[lane16*8 + (hi ? 3 : 1)];
  a1.x = Uo[lane16*8 + (hi ? 6 : 4)];
  a1.y = Uo[lane16*8 + (hi ? 7 : 5)];
  v8f cacc = {};
  cacc = __builtin_amdgcn_wmma_f32_16x16x4_f32(false, a0, false, a0, (short)0, cacc, false, false);
  cacc = __builtin_amdgcn_wmma_f32_16x16x4_f32(false, a1, false, a1, (short)0, cacc, false, false);

  float Eacc = 0.f;
#pragma unroll
  for (int v = 0; v < 8; ++v) {           // C layout: VGPR v -> M = v + hi*8, N = lane16
    int m = v + hi*8;
    Eacc += cacc[v] * H[m*LDA + lane16];
  }
  // remainder: p >= 16 or q >= 16 (68 entries) on VALU
  for (int idx = lane; idx < 68; idx += 32) {
    int p, q;
    if (idx < 36) { p = 16 + (idx/18); q = idx % 18; }
    else          { int t2 = idx - 36; p = t2 >> 1; q = 16 + (t2 & 1); }
    float dsum = 0.f;
#pragma unroll
    for (int t = 0; t < 7; ++t) dsum += Uo[p*8+t]*Uo[q*8+t];
    Eacc += dsum * H[p*LDA+q];
  }
  // wave reduction
#pragma unroll
  for (int m = 16; m >= 1; m >>= 1) Eacc += __shfl_xor(Eacc, m, 32);

  if (lane == 0) {
    float eo = 0.f;
#pragma unroll
    for (int r = 0; r < 7; ++r) eo += sE[w][r];
    float nuc = 49.0f * 0.529177210544f / d;   // NUMBERS = [7,7]
    out[(size_t)BATCHN*324 + (size_t)BATCHN*18 + b] = Eacc + eo + nuc;
  }
}

extern "C" void kernel_launch(void* const* d_in, const int* in_sizes, int n_in,
                              void* d_out, int out_size, void* d_ws, size_t ws_size,
                              hipStream_t stream) {
  (void)in_sizes; (void)n_in; (void)out_size; (void)d_ws; (void)ws_size;
  const float* pos  = (const float*)d_in[0];
  const float* hg   = (const float*)d_in[1];
  const float* emb  = (const float*)d_in[2];
  const float* Win  = (const float*)d_in[3];
  const float* Wf1  = (const float*)d_in[4];
  const float* bf1  = (const float*)d_in[5];
  const float* Wf2  = (const float*)d_in[6];
  const float* bf2  = (const float*)d_in[7];
  const float* Wo1  = (const float*)d_in[8];
  const float* bo1  = (const float*)d_in[9];
  const float* Wo2  = (const float*)d_in[10];
  const float* bo2  = (const float*)d_in[11];
  const float* Wt   = (const float*)d_in[12];
  const float* bt   = (const float*)d_in[13];
  const float* Won1 = (const float*)d_in[14];
  const float* bon1 = (const float*)d_in[15];
  const float* Won2 = (const float*)d_in[16];
  const float* bon2 = (const float*)d_in[17];
  const float* Wof1 = (const float*)d_in[18];
  const float* bof1 = (const float*)d_in[19];
  const float* Wof2 = (const float*)d_in[20];
  const float* bof2 = (const float*)d_in[21];
  const float* Fd   = (const float*)d_in[22];
  const float* mask = (const float*)d_in[23];
  // d_in[24], d_in[25] = idx_i/idx_j, fixed [0,1]/[1,0] in the reference.

  dim3 grid(BATCHN / NWAVES), block(256);
  xschnet_fused<<<grid, block, 0, stream>>>(
      pos, hg, emb, Win, Wf1, bf1, Wf2, bf2, Wo1, bo1, Wo2, bo2,
      Wt, bt, Won1, bon1, Won2, bon2, Wof1, bof1, Wof2, bof2,
      Fd, mask, (float*)d_out);
}